// CO_75849122447516
// MI455X (gfx1250) — compile-verified
//
#include <hip/hip_runtime.h>
#include <hip/hip_bf16.h>
#include <stdint.h>

#ifndef __has_builtin
#define __has_builtin(x) 0
#endif

#define KH   1024
#define KW   1024
#define KHW  (1024 * 1024)
#define KPW  1026                 // padded row stride
#define KPAD (1026 * 1026)        // padded plane elements

typedef __attribute__((ext_vector_type(2))) float v2f;
typedef __attribute__((ext_vector_type(8))) float v8f;

// Typed pointers for the CDNA5 async global->LDS builtins (param 0 is a
// vector-of-int pointer in the global AS, param 1 the LDS AS3 destination).
typedef int async_b64_t  __attribute__((vector_size(8)));
typedef int async_b128_t __attribute__((vector_size(16)));
typedef __attribute__((address_space(1))) async_b64_t*  gbl_b64_ptr;
typedef __attribute__((address_space(3))) async_b64_t*  lds_b64_ptr;
typedef __attribute__((address_space(1))) async_b128_t* gbl_b128_ptr;
typedef __attribute__((address_space(3))) async_b128_t* lds_b128_ptr;

// ---------------------------------------------------------------------------
// Kernel 1: prep — Y plane, colored mask, b = (I,Q)·mask, x0 = b (padded)
// ---------------------------------------------------------------------------
__global__ __launch_bounds__(256) void CO_prep(
    const float* __restrict__ gray, const float* __restrict__ app,
    float* __restrict__ Yp, float* __restrict__ maskp,
    float2* __restrict__ bp, float2* __restrict__ x0 /* padded */) {
  int p = blockIdx.x * 256 + threadIdx.x;
  if (p >= KHW) return;
  const float s = 1.0f / 255.0f;
  float gr = gray[3 * p + 0] * s, gg = gray[3 * p + 1] * s, gb = gray[3 * p + 2] * s;
  float ar = app[3 * p + 0] * s,  ag = app[3 * p + 1] * s,  ab = app[3 * p + 2] * s;
  float diff = fabsf(gr - ar) + fabsf(gg - ag) + fabsf(gb - ab);
  float colored = (diff > 0.01f) ? 1.0f : 0.0f;
  float Y  = 0.3f * gr + 0.59f * gg + 0.11f * gb;
  float ya = 0.3f * ar + 0.59f * ag + 0.11f * ab;
  float iv = 0.74f * (ar - ya) - 0.27f * (ab - ya);
  float qv = 0.48f * (ar - ya) + 0.41f * (ab - ya);
  float2 b = make_float2(colored * iv, colored * qv);
  Yp[p] = Y;
  maskp[p] = colored;
  bp[p] = b;
  int py = p >> 10, px = p & 1023;
  x0[(py + 1) * KPW + (px + 1)] = b;   // interior; halo stays 0 from memset
}

// ---------------------------------------------------------------------------
// Kernel 2: per-pixel 8-neighbor weights (normalized, zeroed where colored),
// packed as two float4s per pixel -> hot loop reads them with 2x b128.
// ---------------------------------------------------------------------------
__global__ __launch_bounds__(256) void CO_weights(
    const float* __restrict__ Yp, const float* __restrict__ maskp,
    float4* __restrict__ wpk /* 2 * KHW float4 */) {
  int p = blockIdx.x * 256 + threadIdx.x;
  if (p >= KHW) return;
  int py = p >> 10, px = p & 1023;
  const int dxr[8] = {-1, -1, -1, 0, 0, 1, 1, 1};
  const int dyc[8] = {-1, 0, 1, -1, 1, -1, 0, 1};
  float Y = Yp[p];
  float nbr[8], val[8];
  float cnt = 1.0f, sum = Y;
#pragma unroll
  for (int k = 0; k < 8; ++k) {
    int ny = py + dxr[k], nx = px + dyc[k];
    bool ok = ((unsigned)ny < (unsigned)KH) && ((unsigned)nx < (unsigned)KW);
    float v = ok ? Yp[ny * KW + nx] : 0.0f;
    nbr[k] = v;
    val[k] = ok ? 1.0f : 0.0f;
    cnt += val[k];
    sum += v * val[k];
  }
  float mean = sum / cnt;
  float var = (Y - mean) * (Y - mean);
#pragma unroll
  for (int k = 0; k < 8; ++k) { float d = nbr[k] - mean; var += d * d * val[k]; }
  var /= cnt;
  float vs = fmaxf(0.6f * var, 2e-6f);
  float w[8], wsum = 0.0f;
#pragma unroll
  for (int k = 0; k < 8; ++k) {
    float d = nbr[k] - Y;
    float e = expf(-(d * d) / vs) * val[k];
    w[k] = e;
    wsum += e;
  }
  float sc = (1.0f - maskp[p]) / wsum;
  wpk[2 * p + 0] = make_float4(w[0] * sc, w[1] * sc, w[2] * sc, w[3] * sc);
  wpk[2 * p + 1] = make_float4(w[4] * sc, w[5] * sc, w[6] * sc, w[7] * sc);
}

// ---------------------------------------------------------------------------
// Kernel 3 (hot, x100): x_new = b + sum_k w_k * shift_k(x).
// Padded src/dst (zero halo) -> no bounds checks. 32x8 tile + halo staged in
// LDS via CDNA5 async global->LDS b128 copies (170 per block, 1 per thread).
// ---------------------------------------------------------------------------
__global__ __launch_bounds__(256) void CO_stencil(
    const float2* __restrict__ xsrc, float2* __restrict__ xdst,
    const float2* __restrict__ bp, const float4* __restrict__ wpk) {
  __shared__ float2 tile[10][34];   // (8+2) x (32+2) halo tile; row = 272B (16-aligned)
  const int ox = blockIdx.x * 32;
  const int oy = blockIdx.y * 8;
  const int tid = threadIdx.x;

#if __has_builtin(__builtin_amdgcn_global_load_async_to_lds_b128)
  // 10 rows x 17 chunks of 16B = 170 async b128 copies, single op per thread.
  if (tid < 170) {
    int ty = tid / 17, tx = (tid - ty * 17) * 2;
    const float2* src = xsrc + (size_t)(oy + ty) * KPW + (ox + tx);
    __builtin_amdgcn_global_load_async_to_lds_b128(
        (gbl_b128_ptr)(uintptr_t)src,
        (lds_b128_ptr)(uintptr_t)&tile[ty][tx],
        0, 0);
  }
#if __has_builtin(__builtin_amdgcn_s_wait_asynccnt)
  __builtin_amdgcn_s_wait_asynccnt(0);
#else
  asm volatile("s_wait_asynccnt 0" ::: "memory");
#endif
#elif __has_builtin(__builtin_amdgcn_global_load_async_to_lds_b64)
  for (int i = tid; i < 340; i += 256) {
    int ty = i / 34, tx = i - ty * 34;
    const float2* src = xsrc + (size_t)(oy + ty) * KPW + (ox + tx);
    __builtin_amdgcn_global_load_async_to_lds_b64(
        (gbl_b64_ptr)(uintptr_t)src,
        (lds_b64_ptr)(uintptr_t)&tile[ty][tx],
        0, 0);
  }
#if __has_builtin(__builtin_amdgcn_s_wait_asynccnt)
  __builtin_amdgcn_s_wait_asynccnt(0);
#else
  asm volatile("s_wait_asynccnt 0" ::: "memory");
#endif
#else
  for (int i = tid; i < 340; i += 256) {
    int ty = i / 34, tx = i - ty * 34;
    tile[ty][tx] = xsrc[(size_t)(oy + ty) * KPW + (ox + tx)];
  }
#endif
  __syncthreads();

  const int lx = tid & 31, ly = tid >> 5;
  const int px = ox + lx, py = oy + ly;
  const int p = py * KW + px;

  float4 wa = wpk[2 * p + 0];       // w0..w3 : one b128
  float4 wb = wpk[2 * p + 1];       // w4..w7 : one b128

  float2 acc = bp[p];
  float2 t;
  t = tile[ly + 0][lx + 0]; acc.x = fmaf(wa.x, t.x, acc.x); acc.y = fmaf(wa.x, t.y, acc.y); // (-1,-1)
  t = tile[ly + 0][lx + 1]; acc.x = fmaf(wa.y, t.x, acc.x); acc.y = fmaf(wa.y, t.y, acc.y); // (-1, 0)
  t = tile[ly + 0][lx + 2]; acc.x = fmaf(wa.z, t.x, acc.x); acc.y = fmaf(wa.z, t.y, acc.y); // (-1, 1)
  t = tile[ly + 1][lx + 0]; acc.x = fmaf(wa.w, t.x, acc.x); acc.y = fmaf(wa.w, t.y, acc.y); // ( 0,-1)
  t = tile[ly + 1][lx + 2]; acc.x = fmaf(wb.x, t.x, acc.x); acc.y = fmaf(wb.x, t.y, acc.y); // ( 0, 1)
  t = tile[ly + 2][lx + 0]; acc.x = fmaf(wb.y, t.x, acc.x); acc.y = fmaf(wb.y, t.y, acc.y); // ( 1,-1)
  t = tile[ly + 2][lx + 1]; acc.x = fmaf(wb.z, t.x, acc.x); acc.y = fmaf(wb.z, t.y, acc.y); // ( 1, 0)
  t = tile[ly + 2][lx + 2]; acc.x = fmaf(wb.w, t.x, acc.x); acc.y = fmaf(wb.w, t.y, acc.y); // ( 1, 1)

  xdst[(size_t)(py + 1) * KPW + (px + 1)] = acc;
}

// ---------------------------------------------------------------------------
// Kernel 4: YIQ -> RGB via V_WMMA_F32_16X16X4_F32 (confirmed lowering).
// ---------------------------------------------------------------------------
__global__ __launch_bounds__(128) void CO_convert(
    const float* __restrict__ Yp, const float2* __restrict__ xfin /* padded */,
    float* __restrict__ out) {
#if __has_builtin(__builtin_amdgcn_wmma_f32_16x16x4_f32)
  const int wave = blockIdx.x * 4 + (threadIdx.x >> 5);
  const int lane = threadIdx.x & 31;
  const int n = lane & 15;          // A: pixel row | B/D: column (channel)
  const bool hi = lane >= 16;

  int p = wave * 16 + n;
  int py = p >> 10, px = p & 1023;
  float Yv = Yp[p];
  float2 iq = xfin[(size_t)(py + 1) * KPW + (px + 1)];

  v2f a;
  a.x = hi ? iq.y : Yv;             // K=2 (Q) | K=0 (Y)
  a.y = hi ? 0.0f : iq.x;           // K=3 (0) | K=1 (I)

  float mc1 = (n == 0) ? 0.9468822170900693f
            : (n == 1) ? -0.27478764629897834f
            : (n == 2) ? -1.1085450346420322f : 0.0f;
  float mc2 = (n == 0) ? 0.6235565819861433f
            : (n == 1) ? -0.6356910791873801f
            : (n == 2) ? 1.7090069284064666f : 0.0f;
  float m0 = (n < 3) ? 1.0f : 0.0f;

  v2f bm;
  bm.x = hi ? mc2 : m0;             // row K=2 | K=0
  bm.y = hi ? 0.0f : mc1;           // row K=3 | K=1

  v8f c = {};
  v8f d = __builtin_amdgcn_wmma_f32_16x16x4_f32(
      false, a, false, bm, (short)0, c, false, false);

  if (n < 3) {
    int base = wave * 16 + (hi ? 8 : 0);
#pragma unroll
    for (int r = 0; r < 8; ++r) {
      float v = fminf(fmaxf(d[r], 0.0f), 1.0f) * 255.0f;
      out[(size_t)(base + r) * 3 + n] = v;
    }
  }
#else
  if (threadIdx.x < 64) {
    int p = blockIdx.x * 64 + threadIdx.x;
    int py = p >> 10, px = p & 1023;
    float y = Yp[p];
    float2 iq = xfin[(size_t)(py + 1) * KPW + (px + 1)];
    float r = y + 0.9468822170900693f  * iq.x + 0.6235565819861433f * iq.y;
    float g = y - 0.27478764629897834f * iq.x - 0.6356910791873801f * iq.y;
    float b = y - 1.1085450346420322f  * iq.x + 1.7090069284064666f * iq.y;
    out[(size_t)p * 3 + 0] = fminf(fmaxf(r, 0.0f), 1.0f) * 255.0f;
    out[(size_t)p * 3 + 1] = fminf(fmaxf(g, 0.0f), 1.0f) * 255.0f;
    out[(size_t)p * 3 + 2] = fminf(fmaxf(b, 0.0f), 1.0f) * 255.0f;
  }
#endif
}

// ---------------------------------------------------------------------------
// Host: workspace layout + launch sequence (all on `stream`, graph-capturable)
// ---------------------------------------------------------------------------
extern "C" void kernel_launch(void* const* d_in, const int* in_sizes, int n_in,
                              void* d_out, int out_size, void* d_ws, size_t ws_size,
                              hipStream_t stream) {
  (void)in_sizes; (void)n_in; (void)out_size; (void)ws_size;
  const float* gray = (const float*)d_in[0];
  const float* app  = (const float*)d_in[1];

  char* ws = (char*)d_ws;
  const size_t OFF_Y  = 0;
  const size_t OFF_W  = OFF_Y + (size_t)KHW * 4;            //  4 MB
  const size_t OFF_B  = OFF_W + (size_t)8 * KHW * 4;        // 32 MB (2 float4/pixel)
  const size_t OFF_M  = OFF_B + (size_t)KHW * 8;            //  8 MB
  const size_t OFF_XA = OFF_M + (size_t)KHW * 4;            //  4 MB
  const size_t OFF_XB = OFF_XA + (size_t)KPAD * 8;          //  ~8.4 MB each

  float*  Yp  = (float*)(ws + OFF_Y);
  float4* wpk = (float4*)(ws + OFF_W);
  float2* bp  = (float2*)(ws + OFF_B);
  float*  mp  = (float*)(ws + OFF_M);
  float2* xA  = (float2*)(ws + OFF_XA);
  float2* xB  = (float2*)(ws + OFF_XB);
  float* out  = (float*)d_out;

  // Zero both padded ping-pong buffers (establishes zero halos permanently).
  (void)hipMemsetAsync(xA, 0, (size_t)2 * KPAD * sizeof(float2), stream);

  CO_prep<<<KHW / 256, 256, 0, stream>>>(gray, app, Yp, mp, bp, xA);
  CO_weights<<<KHW / 256, 256, 0, stream>>>(Yp, mp, wpk);

  float2* src = xA;
  float2* dst = xB;
  for (int it = 0; it < 100; ++it) {
    CO_stencil<<<dim3(KW / 32, KH / 8), 256, 0, stream>>>(src, dst, bp, wpk);
    float2* t = src; src = dst; dst = t;
  }
  // 100 iterations (even) -> result back in xA == src.
  CO_convert<<<KHW / 64, 128, 0, stream>>>(Yp, src, out);
}